// WT_series_decomp_89541478187370
// MI455X (gfx1250) — compile-verified
//
#include <hip/hip_runtime.h>

// ---------------------------------------------------------------------------
// 4-level db4 wavelet trend/season decomposition, one signal per workgroup.
// Whole signal (128 KB) + full pyramid lives in LDS (CDNA5: 320 KB / WGP).
// Global->LDS load uses the CDNA5 Tensor Data Mover (tensor_load_to_lds),
// one 4 KB chunk per wave, synchronized with s_wait_tensorcnt.
// ---------------------------------------------------------------------------

#define NT   1024
#define N0   32768
#define M1   16387
#define M2   8197
#define M3   4102
#define M4   2054
#define HALO 8

// LDS layout (float indices). Each buffer: [8-float zero halo][interior][8-float tail]
#define X_INT 8                       // interior of X, capacity 32768
#define A_INT 32792                   // capacity 16388 (lo1 / t3)
#define B_INT 49196                   // capacity 8198  (lo2 / t2)
#define C_INT 57410                   // capacity 4102  (lo3 / t1)
#define D_INT 61528                   // capacity 2054  (lo4)
#define TOTAL_FLOATS 63590            // ~248.4 KB dynamic LDS

typedef unsigned int tdm_v4u __attribute__((ext_vector_type(4)));
typedef int          tdm_v8i __attribute__((ext_vector_type(8)));
typedef int          tdm_v4i __attribute__((ext_vector_type(4)));

// analysis low-pass: h[t] = DEC_LO[7-t]
#define H0 0.23037781330885523f
#define H1 0.7148465705525415f
#define H2 0.6308807679295904f
#define H3 (-0.02798376941698385f)
#define H4 (-0.18703481171888114f)
#define H5 0.030841381835986965f
#define H6 0.032883011666982945f
#define H7 (-0.010597401784997278f)
// synthesis: g = DEC_LO
#define G0 (-0.010597401784997278f)
#define G1 0.032883011666982945f
#define G2 0.030841381835986965f
#define G3 (-0.18703481171888114f)
#define G4 (-0.02798376941698385f)
#define G5 0.6308807679295904f
#define G6 0.7148465705525415f
#define G7 0.23037781330885523f

extern __shared__ float lds[];

// Stride-2 analysis conv: out[i] = sum_t h[t] * in[2i + t - 6], zero-padded
// via LDS halos. Also zeroes the 8 floats past the logical end so this buffer
// can feed the next level without boundary branches.
__device__ __forceinline__ void afb(const float* __restrict__ in,
                                    float* __restrict__ out, int Mout) {
  for (int i = (int)threadIdx.x; i < Mout; i += NT) {
    const float* p = in + 2 * i - 6;
    out[i] = p[0]*H0 + p[1]*H1 + p[2]*H2 + p[3]*H3 +
             p[4]*H4 + p[5]*H5 + p[6]*H6 + p[7]*H7;
  }
  if (threadIdx.x < HALO) out[Mout + (int)threadIdx.x] = 0.f;
}

// Dilation-2 transposed conv (synthesis): Mout = 2*Min_used - 6.
// even n: {g1,g3,g5,g7} . lo[k..k+3]; odd n: {g0,g2,g4,g6} . lo[k..k+3], k=n>>1
__device__ __forceinline__ void sfb(const float* __restrict__ lo,
                                    float* __restrict__ out, int Mout) {
  for (int n = (int)threadIdx.x; n < Mout; n += NT) {
    int k = n >> 1;
    float l0 = lo[k], l1 = lo[k + 1], l2 = lo[k + 2], l3 = lo[k + 3];
    float e = G1*l0 + G3*l1 + G5*l2 + G7*l3;
    float o = G0*l0 + G2*l1 + G4*l2 + G6*l3;
    out[n] = (n & 1) ? o : e;
  }
}

__global__ __launch_bounds__(NT, 1)
void wt_decomp_kernel(const float* __restrict__ x,
                      float* __restrict__ season,
                      float* __restrict__ trend) {
  const int sig = blockIdx.x;
  const int tid = (int)threadIdx.x;
  float* X = lds + X_INT;
  float* A = lds + A_INT;
  float* B = lds + B_INT;
  float* C = lds + C_INT;
  float* D = lds + D_INT;

  // Zero left halos of all buffers and the right halo of X.
  if (tid < HALO) {
    lds[tid]                 = 0.f;   // X left halo (starts at LDS offset 0)
    lds[X_INT + N0 + tid]    = 0.f;   // X right halo
    lds[A_INT - HALO + tid]  = 0.f;
    lds[B_INT - HALO + tid]  = 0.f;
    lds[C_INT - HALO + tid]  = 0.f;
    lds[D_INT - HALO + tid]  = 0.f;
  }

  // --- TDM async load: each of the 32 waves DMAs its contiguous 1024-float
  // chunk of this block's signal from global memory into LDS (ISA ch.8 D#
  // layout, 1-D tile, data_size=4B, count=1, type=2, no multicast/pad/iterate).
  {
    unsigned wid = (unsigned)__builtin_amdgcn_readfirstlane(tid >> 5);
    unsigned long long ga =
        (unsigned long long)(uintptr_t)(x + (size_t)sig * N0) +
        (unsigned long long)wid * 4096ull;
    unsigned ldsb = (unsigned)(X_INT * 4) + wid * 4096u;     // LDS byte addr
    tdm_v4u g0 = { 1u,                                        // count=1
                   ldsb,                                      // lds_addr
                   (unsigned)ga,                              // global_addr lo
                   (unsigned)((ga >> 32) & 0x01FFFFFFull) | 0x80000000u }; // hi + type=2
    tdm_v8i g1 = { 0x00020000,   // data_size=2 (4 bytes), mask=0
                   0x04000000,   // tensor_dim0[15:0]=1024 in [31:16]
                   0x00010000,   // tensor_dim0 hi=0, tensor_dim1[15:0]=1
                   0x04000000,   // tensor_dim1 hi=0, tile_dim0=1024
                   0,            // tile_dim1=0 (unused), tile_dim2=0
                   1024,         // tensor_dim0_stride lo
                   0, 0 };
    tdm_v4i g2 = { 0, 0, 0, 0 };
    tdm_v4i g3 = { 0, 0, 0, 0 };
    tdm_v8i g4 = { 0, 0, 0, 0, 0, 0, 0, 0 };  // 6-arg toolchain: extra group
    __builtin_amdgcn_tensor_load_to_lds(g0, g1, g2, g3, g4, 0);
    __builtin_amdgcn_s_wait_tensorcnt(0);
  }
  __syncthreads();

  // Analysis cascade (low-pass only; details are discarded by the reference).
  afb(X, A, M1); __syncthreads();   // 32768 -> 16387
  afb(A, B, M2); __syncthreads();   // 16387 -> 8197
  afb(B, C, M3); __syncthreads();   // 8197  -> 4102
  afb(C, D, M4); __syncthreads();   // 4102  -> 2054

  // Synthesis cascade (trims fold into the Mout used at the next step).
  sfb(D, C, 2 * M4   - 6); __syncthreads();   // 2054 -> 4102
  sfb(C, B, 2 * 4102 - 6); __syncthreads();   // 4102 -> 8198 (use 8197)
  sfb(B, A, 2 * 8197 - 6); __syncthreads();   // 8197 -> 16388 (use 16387)

  // Final level fused with season = x - trend; float4 (b128) streaming stores.
  const size_t base = (size_t)sig * N0;
  for (int n0 = tid * 4; n0 < N0; n0 += NT * 4) {
    int k = n0 >> 1;
    float l0 = A[k], l1 = A[k+1], l2 = A[k+2], l3 = A[k+3], l4 = A[k+4];
    float t0 = G1*l0 + G3*l1 + G5*l2 + G7*l3;   // n0   (even)
    float t1 = G0*l0 + G2*l1 + G4*l2 + G6*l3;   // n0+1 (odd)
    float t2 = G1*l1 + G3*l2 + G5*l3 + G7*l4;   // n0+2 (even)
    float t3 = G0*l1 + G2*l2 + G4*l3 + G6*l4;   // n0+3 (odd)
    const float4 xv = *reinterpret_cast<const float4*>(X + n0);
    float4 tr; tr.x = t0; tr.y = t1; tr.z = t2; tr.w = t3;
    float4 se; se.x = xv.x - t0; se.y = xv.y - t1;
               se.z = xv.z - t2; se.w = xv.w - t3;
    *reinterpret_cast<float4*>(season + base + n0) = se;
    *reinterpret_cast<float4*>(trend  + base + n0) = tr;
  }
}

extern "C" void kernel_launch(void* const* d_in, const int* in_sizes, int n_in,
                              void* d_out, int out_size, void* d_ws, size_t ws_size,
                              hipStream_t stream) {
  (void)n_in; (void)out_size; (void)d_ws; (void)ws_size;
  const float* x = (const float*)d_in[0];
  float* out = (float*)d_out;
  const int total = in_sizes[0];          // 32 * 16 * 32768
  const int nsig  = total / N0;           // 512 independent signals
  float* season = out;                    // output tuple: (season, trend)
  float* trend  = out + (size_t)total;
  const size_t shmem = (size_t)TOTAL_FLOATS * sizeof(float);
  hipLaunchKernelGGL(wt_decomp_kernel, dim3(nsig), dim3(NT), shmem, stream,
                     x, season, trend);
}